// GaussianSplatRasterizer_49675591745757
// MI455X (gfx1250) — compile-verified
//
#include <hip/hip_runtime.h>

// ---------------- problem constants (from reference module) ----------------
#define N_PIXC   1024
#define NV_CH    64
#define KW       7
#define KHALF    3
#define MARGIN   4          // kernel2d.shape[-1]//2 + 1
#define CUBE_ELEMS ((long long)NV_CH * N_PIXC * N_PIXC)
#define CUBE_BYTES ((size_t)CUBE_ELEMS * sizeof(float))

// conv tiling: 64x64 output tile per 256-thread block (8 waves)
#define TILE       64
#define RAW_ROWS   70       // TILE + 2*KHALF
#define RAW_COLS   70
#define RAW_STRIDE 70       // contiguous rows so the TDM tile maps 1:1
#define H_ROWS     72       // K padded to 24 per quadrant -> need up to row 71
#define H_STRIDE   65

typedef float v2f __attribute__((ext_vector_type(2)));
typedef float v8f __attribute__((ext_vector_type(8)));
typedef unsigned int u32x4 __attribute__((ext_vector_type(4)));
typedef int i32x4 __attribute__((ext_vector_type(4)));
typedef int i32x8 __attribute__((ext_vector_type(8)));

#if defined(__has_builtin)
#if __has_builtin(__builtin_amdgcn_tensor_load_to_lds) && \
    __has_builtin(__builtin_amdgcn_s_wait_tensorcnt)
#define HAVE_TDM 1
#endif
#endif
#ifndef HAVE_TDM
#define HAVE_TDM 0
#endif

// ---------------------------------------------------------------------------
// Path A kernel 1: scatter fluxes into the delta cube (d_ws).
// Replicates reference masking + flat-index "drop" semantics exactly.
// ---------------------------------------------------------------------------
__global__ void gsplat_scatter_kernel(const float* __restrict__ pos,
                                      const float* __restrict__ vel,
                                      const float* __restrict__ flux,
                                      const float* __restrict__ vaxis,
                                      float* __restrict__ cube, int M) {
  int i = blockIdx.x * blockDim.x + threadIdx.x;
  if (i >= M) return;
  const float FOV_HALF = (float)(0.5 * (N_PIXC - 1) * 0.1);
  const float PIX = 0.1f;
  float ra  = pos[2 * i];
  float dec = pos[2 * i + 1];
  float v0  = vaxis[0];
  float dv  = vaxis[1] - v0;
  int ix = (int)floorf((ra  + FOV_HALF) / PIX);
  int iy = (int)floorf((dec + FOV_HALF) / PIX);
  int iv = (int)floorf((vel[i] - v0) / dv);
  bool ok = (ix >= -MARGIN) && (ix < N_PIXC + MARGIN) &&
            (iy >= -MARGIN) && (iy < N_PIXC + MARGIN) &&
            (iv >= 0) && (iv < NV_CH);
  if (!ok) return;
  long long flat = ((long long)iv * N_PIXC + iy) * (long long)N_PIXC + ix;
  if (flat < 0 || flat >= CUBE_ELEMS) return;   // jax mode="drop"
  atomicAdd(cube + flat, flux[i]);
}

// ---------------------------------------------------------------------------
// Path A kernel 2: fused separable 7x7 conv.
//   * interior tiles: 70x70 f32 sub-tile fetched by the Tensor Data Mover
//     (tensor_load_to_lds + s_wait_tensorcnt), fully in-bounds
//   * border tiles: guarded VALU loads (zero padding == SAME conv)
//   * horizontal 7-tap pass in VALU (x contiguous)
//   * vertical pass as banded-matrix product with V_WMMA_F32_16X16X4_F32:
//       out(16x16) = W(16x24) x H(24x16),  W[m,j] = g1d[j-m] (0 off-band)
// ---------------------------------------------------------------------------
__global__ void gsplat_conv_wmma_kernel(const float* __restrict__ cube,
                                        const float* __restrict__ k2d,
                                        float* __restrict__ out) {
  __shared__ float raw[RAW_ROWS * RAW_STRIDE];   // 19600 B
  __shared__ float hbuf[H_ROWS * H_STRIDE];      // 18720 B
  __shared__ float gsh[8];

  const int tx0 = blockIdx.x * TILE;
  const int ty0 = blockIdx.y * TILE;
  const int ch  = blockIdx.z;
  const int tid = threadIdx.x;

  // g1d[t] = sum_j kernel2d[t][j]  (kernel2d = g1d outer g1d, sum(g1d)=1)
  if (tid < KW) {
    float s = 0.f;
#pragma unroll
    for (int j = 0; j < KW; ++j) s += k2d[tid * KW + j];
    gsh[tid] = s;
  }

  const float* plane = cube + (size_t)ch * N_PIXC * N_PIXC;
  const bool interior = (ty0 >= KHALF) && (ty0 + TILE + KHALF <= N_PIXC) &&
                        (tx0 >= KHALF) && (tx0 + TILE + KHALF <= N_PIXC);

#if HAVE_TDM
  if (interior) {
    // One wave programs the TDM with a 2D D# (CDNA5 ISA).
    if (tid < 32) {
      unsigned long long gaddr =
          (unsigned long long)(const void*)(plane +
              (size_t)(ty0 - KHALF) * N_PIXC + (tx0 - KHALF));
      unsigned lds_off = (unsigned)(unsigned long long)(const void*)raw;
      u32x4 g0;
      g0[0] = 1u;                                   // count=1, user mode
      g0[1] = lds_off;                              // lds_addr
      g0[2] = (unsigned)(gaddr & 0xffffffffu);      // global_addr[31:0]
      g0[3] = (unsigned)((gaddr >> 32) & 0x1ffffffu) | (2u << 30); // type=2
      i32x8 g1;
      g1[0] = (2 << 16);                 // data_size = 4B
      g1[1] = (int)(N_PIXC << 16);       // tensor_dim0 lo16 @ [63:48]
      g1[2] = (int)(N_PIXC << 16);       // dim0 hi16=0 | tensor_dim1 lo16
      g1[3] = (RAW_COLS << 16);          // dim1 hi16=0 | tile_dim0=70
      g1[4] = RAW_ROWS;                  // tile_dim1=70, tile_dim2=0
      g1[5] = N_PIXC;                    // tensor_dim0_stride[31:0]
      g1[6] = 0;
      g1[7] = 0;
      i32x4 g2 = {0, 0, 0, 0};
      i32x4 g3 = {0, 0, 0, 0};
#if __clang_major__ >= 23
      i32x8 g4 = {0, 0, 0, 0, 0, 0, 0, 0};
      __builtin_amdgcn_tensor_load_to_lds(g0, g1, g2, g3, g4, 0);
#else
      __builtin_amdgcn_tensor_load_to_lds(g0, g1, g2, g3, 0);
#endif
      __builtin_amdgcn_s_wait_tensorcnt(0);
    }
  } else
#endif
  {
    // border (or no-TDM) path: guarded loads with zero padding
    for (int i = tid; i < RAW_ROWS * RAW_COLS; i += blockDim.x) {
      int r = i / RAW_COLS, c = i % RAW_COLS;
      int gy = ty0 - KHALF + r;
      int gx = tx0 - KHALF + c;
      float v = 0.f;
      if ((unsigned)gy < (unsigned)N_PIXC && (unsigned)gx < (unsigned)N_PIXC)
        v = plane[(size_t)gy * N_PIXC + gx];
      raw[r * RAW_STRIDE + c] = v;
    }
  }
  __syncthreads();

  // horizontal 7-tap pass: H[r][x] = sum_t g[t]*raw[r][x+t]; rows 70..71 = 0
  for (int i = tid; i < H_ROWS * TILE; i += blockDim.x) {
    int r = i >> 6, x = i & (TILE - 1);
    float s = 0.f;
    if (r < RAW_ROWS) {
#pragma unroll
      for (int t = 0; t < KW; ++t) s += gsh[t] * raw[r * RAW_STRIDE + x + t];
    }
    hbuf[r * H_STRIDE + x] = s;
  }
  __syncthreads();

  // vertical pass via WMMA: 16 quadrants of 16x16, 8 waves -> 2 per wave
  const int wave = tid >> 5;
  const int lane = tid & 31;
  const int n    = lane & 15;   // column of B/C/D; row m of A
  const int half = lane >> 4;

#pragma unroll
  for (int qq = 0; qq < 2; ++qq) {
    const int q  = wave + qq * 8;
    const int qy = q >> 2;
    const int qx = q & 3;
    v8f acc = {};
#pragma unroll
    for (int kk = 0; kk < 24; kk += 4) {
      // A fragment (band weights): VGPR0 holds K = kk+2*half, VGPR1 = K+1
      int j0 = kk + 2 * half;
      int j1 = j0 + 1;
      int d0 = j0 - n;          // m = n (lane&15) for the A matrix
      int d1 = j1 - n;
      v2f a;
      a.x = (d0 >= 0 && d0 < KW) ? gsh[d0] : 0.f;
      a.y = (d1 >= 0 && d1 < KW) ? gsh[d1] : 0.f;
      // B fragment: H rows (qy*16 + k), columns qx*16 + n
      v2f b;
      b.x = hbuf[(qy * 16 + j0) * H_STRIDE + qx * 16 + n];
      b.y = hbuf[(qy * 16 + j1) * H_STRIDE + qx * 16 + n];
      acc = __builtin_amdgcn_wmma_f32_16x16x4_f32(false, a, false, b,
                                                  (short)0, acc, false, false);
    }
    // D layout: lane n (0..15 per half), reg r -> M = 8*half + r
#pragma unroll
    for (int r = 0; r < 8; ++r) {
      int m = half * 8 + r;
      int Y = ty0 + qy * 16 + m;
      int X = tx0 + qx * 16 + n;
      out[((size_t)ch * N_PIXC + Y) * N_PIXC + X] = acc[r];
    }
  }
}

// ---------------------------------------------------------------------------
// Path B (fallback when d_ws can't hold the 256 MiB delta cube):
// splat each point's 7x7 stamp flux*kernel2d directly with global atomics.
// Identical to scatter + SAME-conv for interior points (all inputs are).
// ---------------------------------------------------------------------------
__global__ void gsplat_splat_kernel(const float* __restrict__ pos,
                                    const float* __restrict__ vel,
                                    const float* __restrict__ flux,
                                    const float* __restrict__ vaxis,
                                    const float* __restrict__ k2d,
                                    float* __restrict__ out, int M) {
  int i = blockIdx.x * blockDim.x + threadIdx.x;
  if (i >= M) return;
  const float FOV_HALF = (float)(0.5 * (N_PIXC - 1) * 0.1);
  const float PIX = 0.1f;
  float ra  = pos[2 * i];
  float dec = pos[2 * i + 1];
  float v0  = vaxis[0];
  float dv  = vaxis[1] - v0;
  int ix = (int)floorf((ra  + FOV_HALF) / PIX);
  int iy = (int)floorf((dec + FOV_HALF) / PIX);
  int iv = (int)floorf((vel[i] - v0) / dv);
  bool ok = (ix >= -MARGIN) && (ix < N_PIXC + MARGIN) &&
            (iy >= -MARGIN) && (iy < N_PIXC + MARGIN) &&
            (iv >= 0) && (iv < NV_CH);
  if (!ok) return;
  float f = flux[i];
  float* plane = out + (size_t)iv * N_PIXC * N_PIXC;
#pragma unroll
  for (int dy = 0; dy < KW; ++dy) {
    int y = iy + KHALF - dy;
    if ((unsigned)y >= (unsigned)N_PIXC) continue;
#pragma unroll
    for (int dx = 0; dx < KW; ++dx) {
      int x = ix + KHALF - dx;
      if ((unsigned)x >= (unsigned)N_PIXC) continue;
      atomicAdd(plane + (size_t)y * N_PIXC + x, f * k2d[dy * KW + dx]);
    }
  }
}

// ---------------------------------------------------------------------------
extern "C" void kernel_launch(void* const* d_in, const int* in_sizes, int n_in,
                              void* d_out, int out_size, void* d_ws, size_t ws_size,
                              hipStream_t stream) {
  const float* pos   = (const float*)d_in[0];  // (M,2) interleaved ra/dec
  const float* vel   = (const float*)d_in[1];  // (M,)
  const float* flux  = (const float*)d_in[2];  // (M,)
  const float* vaxis = (const float*)d_in[3];  // (64,)
  const float* k2d   = (const float*)d_in[4];  // (49,)
  float* out = (float*)d_out;                  // (64,1024,1024) f32
  const int M = in_sizes[1];

  if (ws_size >= CUBE_BYTES) {
    // Path A: scatter deltas into ws cube, then fused separable WMMA conv.
    float* cube = (float*)d_ws;
    hipMemsetAsync(cube, 0, CUBE_BYTES, stream);
    gsplat_scatter_kernel<<<(M + 255) / 256, 256, 0, stream>>>(
        pos, vel, flux, vaxis, cube, M);
    dim3 grid(N_PIXC / TILE, N_PIXC / TILE, NV_CH);
    gsplat_conv_wmma_kernel<<<grid, 256, 0, stream>>>(cube, k2d, out);
  } else {
    // Path B: direct 7x7 atomic splat into the output cube.
    hipMemsetAsync(out, 0, CUBE_BYTES, stream);
    gsplat_splat_kernel<<<(M + 255) / 256, 256, 0, stream>>>(
        pos, vel, flux, vaxis, k2d, out, M);
  }
}